// ChargridEmbedding_26714696581699
// MI455X (gfx1250) — compile-verified
//
#include <hip/hip_runtime.h>

// Problem constants (fixed by the reference).
#define B_      4
#define H_      512
#define W_      512
#define N_      256
#define VOCAB_  128
#define D_      64
#define HW_     (H_ * W_)      // 262144 = 2^18
#define LSTRIDE 65             // padded LDS row stride -> bank-conflict-free gather

// Native clang vector types (nontemporal builtins reject HIP_vector_type structs).
typedef int   v4i __attribute__((ext_vector_type(4)));
typedef float v4f __attribute__((ext_vector_type(4)));

// ---------------------------------------------------------------------------
// Phase 2: rasterize boxes into packed priority grid with atomicMax.
// packed = ((n+1) << 8) | token  (token < 128 fits in 8 bits, n+1 <= 256).
// max over packed == "last box wins", default 0 == token 0 (matches reference).
// One block per (b, n) box; area <= 32*32 = 1024 painted pixels.
// atomicMax is commutative -> deterministic result regardless of ordering.
// ---------------------------------------------------------------------------
__global__ __launch_bounds__(256)
void chargrid_raster_kernel(const int* __restrict__ token_ids,
                            const int* __restrict__ boxes,
                            int* __restrict__ grid) {
    const int bn  = blockIdx.x;        // 0 .. B*N-1
    const int b   = bn >> 8;           // N = 256
    const int n   = bn & (N_ - 1);
    const int tok = token_ids[bn] & 0xFF;
    const int w0  = boxes[bn * 4 + 0];
    const int h0  = boxes[bn * 4 + 1];
    const int w1  = boxes[bn * 4 + 2];
    const int h1  = boxes[bn * 4 + 3];
    const int bw  = w1 - w0;
    const int area = bw * (h1 - h0);
    const int packed = ((n + 1) << 8) | tok;

    int* __restrict__ g = grid + b * HW_;
    for (int p = threadIdx.x; p < area; p += 256) {
        const int dy = p / bw;
        const int dx = p - dy * bw;
        atomicMax(&g[(h0 + dy) * W_ + (w0 + dx)], packed);
    }
}

// ---------------------------------------------------------------------------
// Phase 3: embedding gather. Stage the 32KB table into LDS (padded stride 65,
// via gfx1250 async global->LDS), then each thread handles 4 consecutive w
// pixels: one NT b128 load of packed ids, then 64 x (ds_load_2addr pair +
// NT global_store_b128) -> fully-coalesced 256 MB output stream that bypasses
// L2 dirty residency (output > 192 MB L2; written once, never re-read).
// ---------------------------------------------------------------------------
__global__ __launch_bounds__(256)
void chargrid_embed_kernel(const int* __restrict__ grid,
                           const float* __restrict__ emb,
                           float* __restrict__ out) {
    __shared__ float lut[VOCAB_ * LSTRIDE];   // 128*65*4 = 33,280 B

#if __has_builtin(__builtin_amdgcn_global_load_async_to_lds_b32)
    // gfx1250 async DMA path: per-lane global->LDS copy, tracked by ASYNCcnt.
    for (int j = threadIdx.x; j < VOCAB_ * D_; j += 256) {
        const int t = j >> 6;          // vocab row
        const int d = j & (D_ - 1);    // feature
        __builtin_amdgcn_global_load_async_to_lds_b32(
            (int*)(emb + j),
            (int*)(&lut[t * LSTRIDE + d]),
            0, 0);
    }
#if __has_builtin(__builtin_amdgcn_s_wait_asynccnt)
    __builtin_amdgcn_s_wait_asynccnt(0);
#else
    asm volatile("s_wait_asynccnt 0x0" ::: "memory");
#endif
#else
    // Fallback: plain coalesced load + LDS store staging.
    for (int j = threadIdx.x; j < VOCAB_ * D_; j += 256) {
        const int t = j >> 6;
        const int d = j & (D_ - 1);
        lut[t * LSTRIDE + d] = emb[j];
    }
#endif
    __syncthreads();

    // Each thread: one quad of 4 consecutive w pixels.
    const int quad = blockIdx.x * 256 + threadIdx.x;   // 0 .. B*H*W/4 - 1
    const int pix0 = quad << 2;
    const int b    = pix0 >> 18;                        // HW_ = 2^18
    const int rem  = pix0 & (HW_ - 1);
    const int h    = rem >> 9;                          // W_ = 512
    const int w    = rem & (W_ - 1);

    // Packed ids are read exactly once -> non-temporal b128 load.
    const v4i g4 = __builtin_nontemporal_load((const v4i*)(grid + pix0));
    const int t0 = (g4.x & 0xFF) * LSTRIDE;
    const int t1 = (g4.y & 0xFF) * LSTRIDE;
    const int t2 = (g4.z & 0xFF) * LSTRIDE;
    const int t3 = (g4.w & 0xFF) * LSTRIDE;

    float* __restrict__ obase =
        out + (size_t)b * (D_ * HW_) + (size_t)h * W_ + w;

#pragma unroll
    for (int d = 0; d < D_; ++d) {
        v4f v;
        v.x = lut[t0 + d];
        v.y = lut[t1 + d];
        v.z = lut[t2 + d];
        v.w = lut[t3 + d];
        // Streaming output: NT b128 store (write-once, larger than L2).
        __builtin_nontemporal_store(v, (v4f*)(obase + (size_t)d * HW_));
    }
}

// ---------------------------------------------------------------------------
extern "C" void kernel_launch(void* const* d_in, const int* in_sizes, int n_in,
                              void* d_out, int out_size, void* d_ws, size_t ws_size,
                              hipStream_t stream) {
    // Inputs (setup_inputs order): img(f32, unused), token_ids(i32),
    // boxes(i32), emb_table(f32).
    const int*   token_ids = (const int*)d_in[1];
    const int*   boxes     = (const int*)d_in[2];
    const float* emb       = (const float*)d_in[3];
    float*       out       = (float*)d_out;
    int*         grid      = (int*)d_ws;               // B*H*W int32 = 4 MB

    // Phase 1: zero the packed priority grid (capturable, runs every call).
    (void)hipMemsetAsync(grid, 0, (size_t)B_ * HW_ * sizeof(int), stream);

    // Phase 2: paint boxes (one block per box).
    chargrid_raster_kernel<<<B_ * N_, 256, 0, stream>>>(token_ids, boxes, grid);

    // Phase 3: gather + stream 256 MB output.
    chargrid_embed_kernel<<<(B_ * HW_ / 4) / 256, 256, 0, stream>>>(grid, emb, out);
}